// AgentCentricPreProcessing_8383776162287
// MI455X (gfx1250) — compile-verified
//
#include <hip/hip_runtime.h>
#include <cstdint>
#include <cstddef>

typedef long long ll;
typedef unsigned long long u64;
typedef unsigned int u32;
typedef unsigned char u8;

typedef unsigned int v4u __attribute__((ext_vector_type(4)));
typedef int v4i __attribute__((ext_vector_type(4)));
typedef int v8i __attribute__((ext_vector_type(8)));

// ---------------- problem constants ----------------
constexpr int CS = 16;      // scenes
constexpr int CNSTEP = 91;  // time steps
constexpr int CNA = 128;    // agents
constexpr int CNPL = 2048;  // polylines
constexpr int CNPLN = 20;   // nodes per polyline
constexpr int CNTLS = 64;   // traffic-light stops
constexpr int CCUR = 10;    // STEP_CURRENT
constexpr int CNH = 11;     // history steps
constexpr int CNF = 80;     // future steps
constexpr int CNT = 16;     // targets
constexpr int CNO = 64;     // others
constexpr int CNM = 512;    // selected map polylines
constexpr int CNTL = 40;    // selected traffic lights
constexpr int CB = CS * CNT;
constexpr int NCHUNK = 64;              // polylines staged per TDM round
constexpr int SLABF2 = 2 * NCHUNK * CNPLN;  // float2 per slab buffer (pos rows | dir rows)

// ---------------- output offsets (floats, reference return order) ----------------
constexpr ll F_REF_POS  = 0;
constexpr ll F_REF_ROT  = F_REF_POS  + (ll)CB * 2;
constexpr ll F_TGT_POS  = F_REF_ROT  + (ll)CB * 4;
constexpr ll F_TGT_VEL  = F_TGT_POS  + (ll)CB * CNH * 2;
constexpr ll F_TGT_SPD  = F_TGT_VEL  + (ll)CB * CNH * 2;
constexpr ll F_TGT_ACC  = F_TGT_SPD  + (ll)CB * CNH;
constexpr ll F_TGT_YAW  = F_TGT_ACC  + (ll)CB * CNH;
constexpr ll F_TGT_YR   = F_TGT_YAW  + (ll)CB * CNH;
constexpr ll F_TGT_SIZE = F_TGT_YR   + (ll)CB * CNH;
constexpr ll F_GT_POS   = F_TGT_SIZE + (ll)CB * 3;
constexpr ll F_GT_VEL   = F_GT_POS   + (ll)CB * CNF * 2;
constexpr ll F_GT_SPD   = F_GT_VEL   + (ll)CB * CNF * 2;
constexpr ll F_GT_YAW   = F_GT_SPD   + (ll)CB * CNF;
constexpr ll F_OTH_POS  = F_GT_YAW   + (ll)CB * CNF;
constexpr ll F_OTH_VEL  = F_OTH_POS  + (ll)CB * CNO * CNH * 2;
constexpr ll F_OTH_SPD  = F_OTH_VEL  + (ll)CB * CNO * CNH * 2;
constexpr ll F_OTH_ACC  = F_OTH_SPD  + (ll)CB * CNO * CNH;
constexpr ll F_OTH_YAW  = F_OTH_ACC  + (ll)CB * CNO * CNH;
constexpr ll F_OTH_YR   = F_OTH_YAW  + (ll)CB * CNO * CNH;
constexpr ll F_OTH_SIZE = F_OTH_YR   + (ll)CB * CNO * CNH;
constexpr ll F_M_POS    = F_OTH_SIZE + (ll)CB * CNO * 3;
constexpr ll F_M_DIR    = F_M_POS    + (ll)CB * CNM * CNPLN * 2;
constexpr ll F_TL_POS   = F_M_DIR    + (ll)CB * CNM * CNPLN * 2;
constexpr ll F_TL_DIR   = F_TL_POS   + (ll)CB * CNH * CNTL * 2;
constexpr ll B_TIDX     = F_TL_DIR   + (ll)CB * CNH * CNTL * 2;
constexpr ll B_REF_TYPE = B_TIDX     + (ll)CB;
constexpr ll B_REF_ROLE = B_REF_TYPE + (ll)CB * 3;
constexpr ll B_TGT_VALID= B_REF_ROLE + (ll)CB * 3;
constexpr ll B_GT_VALID = B_TGT_VALID+ (ll)CB * CNH;
constexpr ll B_GT_CMD   = B_GT_VALID + (ll)CB * CNF;
constexpr ll B_OTH_VALID= B_GT_CMD   + (ll)CB * 8;
constexpr ll B_OTH_TYPE = B_OTH_VALID+ (ll)CB * CNO * CNH;
constexpr ll B_OTH_ROLE = B_OTH_TYPE + (ll)CB * CNO * 3;
constexpr ll B_M_VALID  = B_OTH_ROLE + (ll)CB * CNO * 3;
constexpr ll B_M_TYPE   = B_M_VALID  + (ll)CB * CNM * CNPLN;
constexpr ll B_TL_VALID = B_M_TYPE   + (ll)CB * CNM * 11;
constexpr ll B_TL_STATE = B_TL_VALID + (ll)CB * CNH * CNTL;

// float2 (8B) views: all 2-vector sections start at even float offsets
static_assert((F_REF_POS | F_TGT_POS | F_TGT_VEL | F_GT_POS | F_GT_VEL |
               F_OTH_POS | F_OTH_VEL | F_M_POS | F_M_DIR | F_TL_POS | F_TL_DIR) % 2 == 0, "");
static_assert(F_REF_ROT % 4 == 0, "");
constexpr ll H_REF_POS = F_REF_POS / 2;
constexpr ll Q_REF_ROT = F_REF_ROT / 4;
constexpr ll H_TGT_POS = F_TGT_POS / 2;
constexpr ll H_TGT_VEL = F_TGT_VEL / 2;
constexpr ll H_GT_POS  = F_GT_POS / 2;
constexpr ll H_GT_VEL  = F_GT_VEL / 2;
constexpr ll H_OTH_POS = F_OTH_POS / 2;
constexpr ll H_OTH_VEL = F_OTH_VEL / 2;
constexpr ll H_M_POS   = F_M_POS / 2;
constexpr ll H_M_DIR   = F_M_DIR / 2;
constexpr ll H_TL_POS  = F_TL_POS / 2;
constexpr ll H_TL_DIR  = F_TL_DIR / 2;

// ---------------- device helpers ----------------
__device__ __forceinline__ u32 fkey(float f) {
  u32 u = __float_as_uint(f);
  return (u & 0x80000000u) ? ~u : (u | 0x80000000u);
}
__device__ __forceinline__ float unfkey(u32 u) {
  return (u & 0x80000000u) ? __uint_as_float(u & 0x7fffffffu) : __uint_as_float(~u);
}

// ascending bitonic sort of n (power of two) u64 keys in LDS; all block threads participate
__device__ __forceinline__ void bitonic_sort(u64* keys, int n) {
  for (int k = 2; k <= n; k <<= 1) {
    for (int j = k >> 1; j > 0; j >>= 1) {
      __syncthreads();
      for (int i = threadIdx.x; i < n; i += blockDim.x) {
        int p = i ^ j;
        if (p > i) {
          bool up = ((i & k) == 0);
          u64 a = keys[i], bk = keys[p];
          if ((a > bk) == up) { keys[i] = bk; keys[p] = a; }
        }
      }
    }
  }
  __syncthreads();
}

__device__ __forceinline__ float local_angle(float rad, float ref) {
  float d = rad - ref;
  return atan2f(sinf(d), cosf(d));
}

// ---------------- TDM gather-mode DMA (CDNA5, ISA §8.7) ----------------
#if defined(__HIP_DEVICE_COMPILE__) && __has_builtin(__builtin_amdgcn_tensor_load_to_lds) && __has_builtin(__builtin_amdgcn_s_wait_tensorcnt)
#define USE_TDM 1
#else
#define USE_TDM 0
#endif

#if USE_TDM
__device__ __forceinline__ void tdm_gather16(v4u g0, v8i g1, v4i g2, v4i g3) {
#if __clang_major__ >= 23
  v8i z8 = {0, 0, 0, 0, 0, 0, 0, 0};
  __builtin_amdgcn_tensor_load_to_lds(g0, g1, g2, g3, z8, 0);
#else
  __builtin_amdgcn_tensor_load_to_lds(g0, g1, g2, g3, 0);
#endif
}
#endif

// ================= kernel 1: per-scene target selection + reference frame =================
__global__ __launch_bounds__(128)
void k1_select(const u8* agent_valid, const float* agent_pos, const float* agent_yawb,
               const u8* agent_type, const u8* agent_role, float* out, float* ws) {
  int s = blockIdx.x;
  __shared__ u64 keys[CNA];
  int a = threadIdx.x;
  if (a < CNA) {
    float w = (float)(agent_role[((ll)s * CNA + a) * 3 + 0] +
                      agent_role[((ll)s * CNA + a) * 3 + 1] +
                      agent_role[((ll)s * CNA + a) * 3 + 2]) +
              (float)agent_valid[((ll)s * CNSTEP + CCUR) * CNA + a];
    // top_k descending, ties -> lower index first: ascending sort of (~fkey(w), idx)
    keys[a] = ((u64)(~fkey(w)) << 32) | (u32)a;
  }
  bitonic_sort(keys, CNA);
  if (a < CNT) {
    int t = a;
    int ai = (int)(keys[t] & 0xffffffffu);
    ll b = (ll)s * CNT + t;
    float2 rp = ((const float2*)agent_pos)[((ll)s * CNSTEP + CCUR) * CNA + ai];
    float yaw = agent_yawb[((ll)s * CNSTEP + CCUR) * CNA + ai];
    float c = cosf(yaw), sn = sinf(yaw);
    ((float2*)out)[H_REF_POS + b] = rp;
    float4 rot; rot.x = c; rot.y = -sn; rot.z = sn; rot.w = c;
    ((float4*)out)[Q_REF_ROT + b] = rot;
    out[B_TIDX + b] = (float)ai;
    for (int q = 0; q < 3; q++) {
      out[B_REF_TYPE + b * 3 + q] = (float)agent_type[((ll)s * CNA + ai) * 3 + q];
      out[B_REF_ROLE + b * 3 + q] = (float)agent_role[((ll)s * CNA + ai) * 3 + q];
    }
    float* w8 = ws + b * 8;
    w8[0] = rp.x; w8[1] = rp.y; w8[2] = c; w8[3] = sn;
    ((int*)w8)[4] = ai;
    w8[5] = yaw;
  }
}

// ================= kernel 2: per-(s,t) target history + GT future =================
__global__ __launch_bounds__(128)
void k2_target(const u8* agent_valid, const float* agent_pos, const float* agent_vel,
               const float* agent_spd, const float* agent_acc, const float* agent_yawb,
               const float* agent_yr, const float* agent_size, const u8* agent_cmd,
               float* out, const float* ws) {
  int b = blockIdx.x;
  int s = b / CNT;
  const float* w8 = ws + (ll)b * 8;
  float rx = w8[0], ry = w8[1], c = w8[2], sn = w8[3], ryaw = w8[5];
  int ai = ((const int*)w8)[4];
  const float2* pos2 = (const float2*)agent_pos;
  const float2* vel2 = (const float2*)agent_vel;
  float2* out2 = (float2*)out;

  for (int k = threadIdx.x; k < CNSTEP; k += blockDim.x) {
    ll idx = (ll)(s * CNSTEP + k) * CNA + ai;
    float2 p = pos2[idx], v = vel2[idx];
    float dx = p.x - rx, dy = p.y - ry;
    float2 lp; lp.x = dx * c + dy * sn; lp.y = -dx * sn + dy * c;
    float2 lv; lv.x = v.x * c + v.y * sn; lv.y = -v.x * sn + v.y * c;
    float spd = agent_spd[idx];
    float lyaw = local_angle(agent_yawb[idx], ryaw);
    float vld = (float)agent_valid[(ll)(s * CNSTEP + k) * CNA + ai];
    if (k < CNH) {
      ll o = (ll)b * CNH + k;
      out[B_TGT_VALID + o] = vld;
      out2[H_TGT_POS + o] = lp;
      out2[H_TGT_VEL + o] = lv;
      out[F_TGT_SPD + o] = spd;
      out[F_TGT_ACC + o] = agent_acc[idx];
      out[F_TGT_YAW + o] = lyaw;
      out[F_TGT_YR + o] = agent_yr[idx];
    } else {
      ll o = (ll)b * CNF + (k - CNH);
      out[B_GT_VALID + o] = vld;
      out2[H_GT_POS + o] = lp;
      out2[H_GT_VEL + o] = lv;
      out[F_GT_SPD + o] = spd;
      out[F_GT_YAW + o] = lyaw;
    }
  }
  if (threadIdx.x < 3)
    out[F_TGT_SIZE + (ll)b * 3 + threadIdx.x] = agent_size[((ll)s * CNA + ai) * 3 + threadIdx.x];
  if (threadIdx.x < 8)
    out[B_GT_CMD + (ll)b * 8 + threadIdx.x] = (float)agent_cmd[((ll)s * CNA + ai) * 8 + threadIdx.x];
}

// ================= kernel 3: per-(s,t) other-agent top-64 + gather =================
__global__ __launch_bounds__(128)
void k3_others(const u8* agent_valid, const float* agent_pos, const float* agent_vel,
               const float* agent_spd, const float* agent_acc, const float* agent_yawb,
               const float* agent_yr, const u8* agent_type, const u8* agent_role,
               const float* agent_size, float* out, const float* ws) {
  int b = blockIdx.x;
  int s = b / CNT;
  const float* w8 = ws + (ll)b * 8;
  float rx = w8[0], ry = w8[1], c = w8[2], sn = w8[3], ryaw = w8[5];
  int ai = ((const int*)w8)[4];
  const float2* pos2 = (const float2*)agent_pos;
  const float2* vel2 = (const float2*)agent_vel;
  float2* out2 = (float2*)out;

  __shared__ u64 keys[CNA];
  int a = threadIdx.x;
  if (a < CNA) {
    float2 p = pos2[(ll)(s * CNSTEP + CCUR) * CNA + a];
    float dx = p.x - rx, dy = p.y - ry;
    bool v = agent_valid[(ll)(s * CNSTEP + CCUR) * CNA + a] && (a != ai);
    float od = v ? sqrtf(dx * dx + dy * dy) : __builtin_inff();
    keys[a] = ((u64)fkey(od) << 32) | (u32)a;
  }
  bitonic_sort(keys, CNA);

  for (int i = threadIdx.x; i < CNO * CNH; i += blockDim.x) {
    int o = i / CNH, k = i % CNH;
    int ao = (int)(keys[o] & 0xffffffffu);
    float dist = unfkey((u32)(keys[o] >> 32));
    ll idx = (ll)(s * CNSTEP + k) * CNA + ao;
    __builtin_prefetch(&agent_pos[idx * 2], 0, 0);
    float2 p = pos2[idx], v = vel2[idx];
    float dx = p.x - rx, dy = p.y - ry;
    ll oo = ((ll)b * CNO + o) * CNH + k;
    bool vld = agent_valid[(ll)(s * CNSTEP + k) * CNA + ao] && (dist < 2000.0f);
    out[B_OTH_VALID + oo] = vld ? 1.0f : 0.0f;
    float2 lp; lp.x = dx * c + dy * sn; lp.y = -dx * sn + dy * c;
    float2 lv; lv.x = v.x * c + v.y * sn; lv.y = -v.x * sn + v.y * c;
    out2[H_OTH_POS + oo] = lp;
    out2[H_OTH_VEL + oo] = lv;
    out[F_OTH_SPD + oo] = agent_spd[idx];
    out[F_OTH_ACC + oo] = agent_acc[idx];
    out[F_OTH_YAW + oo] = local_angle(agent_yawb[idx], ryaw);
    out[F_OTH_YR + oo] = agent_yr[idx];
  }
  for (int o = threadIdx.x; o < CNO; o += blockDim.x) {
    int ao = (int)(keys[o] & 0xffffffffu);
    for (int q = 0; q < 3; q++) {
      out[B_OTH_TYPE + ((ll)b * CNO + o) * 3 + q] = (float)agent_type[((ll)s * CNA + ao) * 3 + q];
      out[B_OTH_ROLE + ((ll)b * CNO + o) * 3 + q] = (float)agent_role[((ll)s * CNA + ao) * 3 + q];
      out[F_OTH_SIZE + ((ll)b * CNO + o) * 3 + q] = agent_size[((ll)s * CNA + ao) * 3 + q];
    }
  }
}

// ================= kernel 4: per-(s,t) map top-512, double-buffered TDM gather =================
__global__ __launch_bounds__(256)
void k4_map(const u8* map_valid, const u8* map_type, const float* map_pos,
            const float* map_dir, float* out, const float* ws) {
  int b = blockIdx.x;
  int s = b / CNT;
  const float* w8 = ws + (ll)b * 8;
  float rx = w8[0], ry = w8[1], c = w8[2], sn = w8[3];
  const float2* mpos2 = (const float2*)map_pos;
  const float2* mdir2 = (const float2*)map_dir;
  float2* out2 = (float2*)out;

  __shared__ u64 mkeys[CNPL];            // 16 KB
  __shared__ float2 slab[2][SLABF2];     // 2 x 20 KB: [pos rows 64x20 | dir rows 64x20]

  // phase 1: streaming distance pass (node 0) with prefetch ahead
  for (int pl = threadIdx.x; pl < CNPL; pl += blockDim.x) {
    if (pl + (int)blockDim.x < CNPL)
      __builtin_prefetch(&map_pos[(((ll)s * CNPL + pl + blockDim.x) * CNPLN) * 2], 0, 0);
    float2 p0 = mpos2[((ll)s * CNPL + pl) * CNPLN];
    float dx = p0.x - rx, dy = p0.y - ry;
    float d = map_valid[((ll)s * CNPL + pl) * CNPLN] ? sqrtf(dx * dx + dy * dy)
                                                     : __builtin_inff();
    mkeys[pl] = ((u64)fkey(d) << 32) | (u32)pl;
  }
  bitonic_sort(mkeys, CNPL);

  const ll gbase = (ll)s * CNPL * CNPLN * 2;  // float offset of this scene's map tensors

#if USE_TDM
  // Issue 8 gather-mode TDM descriptors (16 rows x 160B each) for one chunk.
  // Wave-level DMA: TDM ignores EXEC; tracked with TENSORcnt, in-order per wave.
  auto issue_chunk = [&](int c0, int buf) {
    u32 lds_pos = (u32)(size_t)(void*)&slab[buf][0];
    u32 lds_dir = (u32)(size_t)(void*)&slab[buf][NCHUNK * CNPLN];
    u64 ga_pos = (u64)(size_t)(const void*)(map_pos + gbase);
    u64 ga_dir = (u64)(size_t)(const void*)(map_dir + gbase);
    // D# group1: data_size=8B, tensor_dim0=20, tensor_dim1=2048,
    // tile_dim0=20, tile_dim1(#indices)=16, tensor_dim0_stride=20
    v8i g1v = { (3 << 16), (CNPLN << 16), (CNPL << 16), (CNPLN << 16),
                16, CNPLN, 0, 0 };
    for (int g = 0; g < 4; g++) {
      int base = c0 + g * 16;
      u32 p0 = (u32)(mkeys[base + 0] & 0xffffu) | ((u32)(mkeys[base + 1] & 0xffffu) << 16);
      u32 p1 = (u32)(mkeys[base + 2] & 0xffffu) | ((u32)(mkeys[base + 3] & 0xffffu) << 16);
      u32 p2 = (u32)(mkeys[base + 4] & 0xffffu) | ((u32)(mkeys[base + 5] & 0xffffu) << 16);
      u32 p3 = (u32)(mkeys[base + 6] & 0xffffu) | ((u32)(mkeys[base + 7] & 0xffffu) << 16);
      u32 p4 = (u32)(mkeys[base + 8] & 0xffffu) | ((u32)(mkeys[base + 9] & 0xffffu) << 16);
      u32 p5 = (u32)(mkeys[base +10] & 0xffffu) | ((u32)(mkeys[base +11] & 0xffffu) << 16);
      u32 p6 = (u32)(mkeys[base +12] & 0xffffu) | ((u32)(mkeys[base +13] & 0xffffu) << 16);
      u32 p7 = (u32)(mkeys[base +14] & 0xffffu) | ((u32)(mkeys[base +15] & 0xffffu) << 16);
      v4i g2v = { (int)p0, (int)p1, (int)p2, (int)p3 };
      v4i g3v = { (int)p4, (int)p5, (int)p6, (int)p7 };
      // D# group0: gather_mode=1, 16-bit indices, count=1, type=2 ("image")
      v4u g0p = { 0x80000001u, lds_pos + (u32)g * 16 * 160,
                  (u32)ga_pos, ((u32)(ga_pos >> 32) & 0x01ffffffu) | (2u << 30) };
      v4u g0d = { 0x80000001u, lds_dir + (u32)g * 16 * 160,
                  (u32)ga_dir, ((u32)(ga_dir >> 32) & 0x01ffffffu) | (2u << 30) };
      tdm_gather16(g0p, g1v, g2v, g3v);
      tdm_gather16(g0d, g1v, g2v, g3v);
    }
  };
  if (threadIdx.x < 32) issue_chunk(0, 0);   // prologue: DMA for chunk 0
#endif

  for (int ci = 0; ci < CNM / NCHUNK; ci++) {
    int c0 = ci * NCHUNK;
    int cur = ci & 1;
#if USE_TDM
    if (threadIdx.x < 32) {
      if (ci + 1 < CNM / NCHUNK) {
        issue_chunk(c0 + NCHUNK, cur ^ 1);          // prefetch next chunk's DMA
        __builtin_amdgcn_s_wait_tensorcnt(8);       // in-order: chunk ci resident
      } else {
        __builtin_amdgcn_s_wait_tensorcnt(0);
      }
    }
    __syncthreads();
#else
    for (int i = threadIdx.x; i < NCHUNK * CNPLN; i += blockDim.x) {
      int rl = i / CNPLN, nd = i % CNPLN;
      int pl = (int)(mkeys[c0 + rl] & 0xffffffffu);
      slab[cur][rl * CNPLN + nd] = mpos2[((ll)s * CNPL + pl) * CNPLN + nd];
      slab[cur][NCHUNK * CNPLN + rl * CNPLN + nd] = mdir2[((ll)s * CNPL + pl) * CNPLN + nd];
    }
    __syncthreads();
#endif
    // transform + write chunk ci from slab[cur]
    for (int i = threadIdx.x; i < NCHUNK * CNPLN; i += blockDim.x) {
      int rl = i / CNPLN, nd = i % CNPLN;
      int r = c0 + rl;
      int pl = (int)(mkeys[r] & 0xffffffffu);
      float dist = unfkey((u32)(mkeys[r] >> 32));
      float2 p = slab[cur][rl * CNPLN + nd];
      float2 u = slab[cur][NCHUNK * CNPLN + rl * CNPLN + nd];
      float ax = p.x - rx, ay = p.y - ry;
      ll o = ((ll)b * CNM + r) * CNPLN + nd;
      float2 lp; lp.x = ax * c + ay * sn; lp.y = -ax * sn + ay * c;
      float2 lu; lu.x = u.x * c + u.y * sn; lu.y = -u.x * sn + u.y * c;
      out2[H_M_POS + o] = lp;
      out2[H_M_DIR + o] = lu;
      bool mv = map_valid[((ll)s * CNPL + pl) * CNPLN + nd] && (dist < 3000.0f);
      out[B_M_VALID + o] = mv ? 1.0f : 0.0f;
    }
    for (int i = threadIdx.x; i < NCHUNK * 11; i += blockDim.x) {
      int rl = i / 11, q = i % 11;
      int r = c0 + rl;
      int pl = (int)(mkeys[r] & 0xffffffffu);
      out[B_M_TYPE + ((ll)b * CNM + r) * 11 + q] =
          (float)map_type[((ll)s * CNPL + pl) * 11 + q];
    }
    __syncthreads();  // buffer-reuse barrier before next iteration's DMA issue
  }
}

// ================= kernel 5: per-(s,t,h) traffic-light top-40 =================
__global__ __launch_bounds__(64)
void k5_tl(const u8* tl_valid, const u8* tl_state, const float* tl_pos,
           const float* tl_dir, float* out, const float* ws) {
  int blk = blockIdx.x;
  int h = blk % CNH;
  int b = blk / CNH;
  int s = b / CNT;
  const float* w8 = ws + (ll)b * 8;
  float rx = w8[0], ry = w8[1], c = w8[2], sn = w8[3];
  const float2* tp2 = (const float2*)tl_pos;
  const float2* td2 = (const float2*)tl_dir;
  float2* out2 = (float2*)out;

  __shared__ u64 keys[CNTLS];
  int i = threadIdx.x;
  {
    ll idx = (ll)(s * CNSTEP + h) * CNTLS + i;
    float2 p = tp2[idx];
    float dx = p.x - rx, dy = p.y - ry;
    float d = tl_valid[idx] ? sqrtf(dx * dx + dy * dy) : __builtin_inff();
    keys[i] = ((u64)fkey(d) << 32) | (u32)i;
  }
  bitonic_sort(keys, CNTLS);

  if (i < CNTL) {
    int j = (int)(keys[i] & 0xffffffffu);
    float dist = unfkey((u32)(keys[i] >> 32));
    ll idx = (ll)(s * CNSTEP + h) * CNTLS + j;
    ll o = ((ll)b * CNH + h) * CNTL + i;
    bool vld = tl_valid[idx] && (dist < 2000.0f);
    out[B_TL_VALID + o] = vld ? 1.0f : 0.0f;
    for (int q = 0; q < 5; q++)
      out[B_TL_STATE + o * 5 + q] = (float)tl_state[idx * 5 + q];
    float2 p = tp2[idx], u = td2[idx];
    float dx = p.x - rx, dy = p.y - ry;
    float2 lp; lp.x = dx * c + dy * sn; lp.y = -dx * sn + dy * c;
    float2 lu; lu.x = u.x * c + u.y * sn; lu.y = -u.x * sn + u.y * c;
    out2[H_TL_POS + o] = lp;
    out2[H_TL_DIR + o] = lu;
  }
}

// ================= host launcher =================
extern "C" void kernel_launch(void* const* d_in, const int* in_sizes, int n_in,
                              void* d_out, int out_size, void* d_ws, size_t ws_size,
                              hipStream_t stream) {
  (void)in_sizes; (void)n_in; (void)out_size; (void)ws_size;
  const u8*    agent_valid = (const u8*)d_in[0];
  const float* agent_pos   = (const float*)d_in[1];
  const float* agent_vel   = (const float*)d_in[2];
  const float* agent_spd   = (const float*)d_in[3];
  const float* agent_acc   = (const float*)d_in[4];
  const float* agent_yawb  = (const float*)d_in[5];
  const float* agent_yr    = (const float*)d_in[6];
  const u8*    agent_type  = (const u8*)d_in[7];
  const u8*    agent_role  = (const u8*)d_in[8];
  const float* agent_size  = (const float*)d_in[9];
  const u8*    agent_cmd   = (const u8*)d_in[10];
  const u8*    map_valid   = (const u8*)d_in[11];
  const u8*    map_type    = (const u8*)d_in[12];
  const float* map_pos     = (const float*)d_in[13];
  const float* map_dir     = (const float*)d_in[14];
  const u8*    tl_valid    = (const u8*)d_in[15];
  const u8*    tl_state    = (const u8*)d_in[16];
  const float* tl_pos      = (const float*)d_in[17];
  const float* tl_dir      = (const float*)d_in[18];
  float* out = (float*)d_out;
  float* ws  = (float*)d_ws;

  k1_select<<<CS, 128, 0, stream>>>(agent_valid, agent_pos, agent_yawb,
                                    agent_type, agent_role, out, ws);
  k2_target<<<CB, 128, 0, stream>>>(agent_valid, agent_pos, agent_vel, agent_spd,
                                    agent_acc, agent_yawb, agent_yr, agent_size,
                                    agent_cmd, out, ws);
  k3_others<<<CB, 128, 0, stream>>>(agent_valid, agent_pos, agent_vel, agent_spd,
                                    agent_acc, agent_yawb, agent_yr, agent_type,
                                    agent_role, agent_size, out, ws);
  k4_map<<<CB, 256, 0, stream>>>(map_valid, map_type, map_pos, map_dir, out, ws);
  k5_tl<<<CB * CNH, 64, 0, stream>>>(tl_valid, tl_state, tl_pos, tl_dir, out, ws);
}